// GaussianMixture_36996848288358
// MI455X (gfx1250) — compile-verified
//
#include <hip/hip_runtime.h>
#include <math.h>

// CDNA5 / gfx1250 GMM negative log-likelihood.
// Hot loop: V_WMMA_F32_16X16X4_F32 (f32 matrix core) computing M_k*X_tile^T - b_k
// (b folded into the C accumulator) for 16-point tiles, K=32 clusters, with
// M_k = chol(Sigma_k)^-1 precomputed on-device. Wave32 layouts per ISA 7.12.2.
// k-loop unroll capped at 2 to stay under 256 VGPRs (no spills / vgpr-msb mode).

typedef __attribute__((ext_vector_type(2))) float v2f;
typedef __attribute__((ext_vector_type(4))) float v4f;
typedef __attribute__((ext_vector_type(8))) float v8f;

#define NPTS   262144
#define DDIM   16
#define KCL    32

// workspace layout (float offsets)
#define WS_M    0        // 32 * 256 : M_k = L_k^{-1}, row-major [k][row][col]
#define WS_B    8192     // 32 * 16  : -b_k = -(M_k * mu_k)   (stored negated)
#define WS_C    8704     // 32       : c'_k = logsoftmax(w)_k - lognorm_k + C0
#define WS_C0   8736     // 1        : C0 shift
#define WS_PART 8768     // 256      : per-block partial sums

#define WPB 8            // waves per block
#define TPW 8            // 16-point tiles per wave
#define NBLK 256         // 256 blk * 8 waves * 8 tiles * 16 pts = 262144

// ---------------------------------------------------------------------------
// Setup: one wave, lane k owns cluster k.
// Sigma = C C^T + I ; L = chol(Sigma) ; M = L^{-1} ; b = M mu (stored as -b) ;
// c_k = logsoftmax(w)_k - 0.5*(D log 2pi + logdet) ; C0 = -max c_k.
// ---------------------------------------------------------------------------
__global__ __launch_bounds__(32) void gmm_setup(const float* __restrict__ means,
                                                const float* __restrict__ chols,
                                                const float* __restrict__ weights,
                                                float* __restrict__ ws) {
  __shared__ float S[KCL * 256];   // per-cluster Sigma -> in-place chol L
  const int k = threadIdx.x;       // 0..31, one cluster per lane (wave32)
  const float* Ck = chols + k * 256;
  float* Sk = S + k * 256;

  // Sigma lower triangle = C C^T + I
  for (int i = 0; i < DDIM; ++i)
    for (int j = 0; j <= i; ++j) {
      float s = (i == j) ? 1.0f : 0.0f;
      for (int t = 0; t < DDIM; ++t)
        s = fmaf(Ck[i * 16 + t], Ck[j * 16 + t], s);
      Sk[i * 16 + j] = s;
    }

  // Cholesky in place (lower), accumulate logdet
  float logdet = 0.0f;
  for (int j = 0; j < DDIM; ++j) {
    float d = Sk[j * 16 + j];
    for (int t = 0; t < j; ++t) d = fmaf(-Sk[j * 16 + t], Sk[j * 16 + t], d);
    d = sqrtf(d);
    Sk[j * 16 + j] = d;
    logdet += 2.0f * logf(d);
    float inv = 1.0f / d;
    for (int i = j + 1; i < DDIM; ++i) {
      float s = Sk[i * 16 + j];
      for (int t = 0; t < j; ++t) s = fmaf(-Sk[i * 16 + t], Sk[j * 16 + t], s);
      Sk[i * 16 + j] = s * inv;
    }
  }

  // M = L^{-1} column by column; b = M mu accumulated on the fly
  float* Mg = ws + WS_M + k * 256;
  for (int i = 0; i < 256; ++i) Mg[i] = 0.0f;
  float bacc[DDIM];
  for (int i = 0; i < DDIM; ++i) bacc[i] = 0.0f;
  for (int j = 0; j < DDIM; ++j) {
    float col[DDIM];
    col[j] = 1.0f / Sk[j * 16 + j];
    for (int i = j + 1; i < DDIM; ++i) {
      float s = 0.0f;
      for (int t = j; t < i; ++t) s = fmaf(Sk[i * 16 + t], col[t], s);
      col[i] = -s / Sk[i * 16 + i];
    }
    const float mj = means[k * 16 + j];
    for (int i = j; i < DDIM; ++i) {
      Mg[i * 16 + j] = col[i];
      bacc[i] = fmaf(col[i], mj, bacc[i]);
    }
  }
  for (int i = 0; i < DDIM; ++i) ws[WS_B + k * 16 + i] = -bacc[i];  // negated

  // log-softmax of weights across the wave (lane == cluster)
  float wv = weights[k];
  float wmax = wv;
  for (int off = 16; off >= 1; off >>= 1)
    wmax = fmaxf(wmax, __shfl_xor(wmax, off, 32));
  float esum = __expf(wv - wmax);
  for (int off = 16; off >= 1; off >>= 1)
    esum += __shfl_xor(esum, off, 32);
  const float LOG2PI = 1.8378770664093453f;
  float c = (wv - wmax - logf(esum)) - 0.5f * (16.0f * LOG2PI + logdet);
  float cmax = c;
  for (int off = 16; off >= 1; off >>= 1)
    cmax = fmaxf(cmax, __shfl_xor(cmax, off, 32));
  float C0 = -cmax;  // shift so max component coefficient is exp(0)=1
  ws[WS_C + k] = c + C0;
  if (k == 0) ws[WS_C0] = C0;
}

// ---------------------------------------------------------------------------
// Main: each wave owns TPW 16-point tiles. For each tile and cluster:
//   acc(16x16) = M_k(16x16) * Xtile^T(16x16) - b_k  via 4x V_WMMA_F32_16X16X4_F32
//   (-b_k is the initial C accumulator: C vgpr r holds out-dim r+8*hi for every
//    lane, which is point-independent -> broadcast of -b)
//   maha[p] = || acc[:,p] ||^2 ; pdf[p] += exp(c'_k - 0.5 maha)
// A-operand layout (16x4 f32): lane = row (&15), K = 2*(lane>>4) + vgpr.
// B-operand layout (4x16 f32): lane = col (&15), K = 2*(lane>>4) + vgpr.
// C/D layout: vgpr r, lane l -> out-dim = r + 8*(l>>4), point = l&15.
// ---------------------------------------------------------------------------
__global__ __launch_bounds__(256) void gmm_main(const float* __restrict__ X,
                                                float* __restrict__ ws) {
  __shared__ float shM[KCL * 256];
  __shared__ float shNB[KCL * 16];   // holds -b
  __shared__ float shC[KCL];
  __shared__ float shW[WPB];

  const int tid = threadIdx.x;
  for (int i = tid; i < KCL * 256; i += 256) shM[i] = ws[WS_M + i];
  for (int i = tid; i < KCL * 16; i += 256) shNB[i] = ws[WS_B + i];
  if (tid < KCL) shC[tid] = ws[WS_C + tid];
  __syncthreads();

  const int wv = tid >> 5;
  const int lane = tid & 31;
  const int pl = lane & 15;       // point within tile / matrix row
  const int hi = lane >> 4;       // half-wave select
  const int gw = blockIdx.x * WPB + wv;

  float pdfacc = 0.0f;
#pragma clang loop unroll(disable)
  for (int t = 0; t < TPW; ++t) {
    const int tile = gw * TPW + t;
    const float* xp = X + (size_t)tile * 256 + pl * 16 + 2 * hi;
    if (t + 1 < TPW)  // global_prefetch_b8, keep near (WGP scope)
      __builtin_prefetch((const void*)(xp + 256), 0, 3);

    // B operand: X tile, K-slabs of 4 dims each
    v2f B0 = *(const v2f*)(xp);
    v2f B1 = *(const v2f*)(xp + 4);
    v2f B2 = *(const v2f*)(xp + 8);
    v2f B3 = *(const v2f*)(xp + 12);

#pragma clang loop unroll_count(2)
    for (int k = 0; k < KCL; ++k) {
      const float* Mk = shM + k * 256 + pl * 16 + 2 * hi;
      v2f A0 = *(const v2f*)(Mk);
      v2f A1 = *(const v2f*)(Mk + 4);
      v2f A2 = *(const v2f*)(Mk + 8);
      v2f A3 = *(const v2f*)(Mk + 12);

      // initial accumulator = -b_k (broadcast across points)
      const float* nbk = shNB + k * 16 + 8 * hi;
      v4f nb0 = *(const v4f*)(nbk);
      v4f nb1 = *(const v4f*)(nbk + 4);
      v8f acc = {nb0.x, nb0.y, nb0.z, nb0.w, nb1.x, nb1.y, nb1.z, nb1.w};

      acc = __builtin_amdgcn_wmma_f32_16x16x4_f32(false, A0, false, B0,
                                                  (short)0, acc, false, false);
      acc = __builtin_amdgcn_wmma_f32_16x16x4_f32(false, A1, false, B1,
                                                  (short)0, acc, false, false);
      acc = __builtin_amdgcn_wmma_f32_16x16x4_f32(false, A2, false, B2,
                                                  (short)0, acc, false, false);
      acc = __builtin_amdgcn_wmma_f32_16x16x4_f32(false, A3, false, B3,
                                                  (short)0, acc, false, false);

      float m = 0.0f;
#pragma unroll
      for (int r = 0; r < 8; ++r) m = fmaf(acc[r], acc[r], m);
      m += __shfl_xor(m, 16, 32);           // merge the two 8-dim halves
      pdfacc += __expf(fmaf(-0.5f, m, shC[k]));  // arg <= 0 by C0 shift
    }
  }

  // wave reduction; each point appears in two lanes (pl and pl+16)
#pragma unroll
  for (int off = 16; off >= 1; off >>= 1)
    pdfacc += __shfl_xor(pdfacc, off, 32);
  pdfacc *= 0.5f;
  if (lane == 0) shW[wv] = pdfacc;
  __syncthreads();
  if (tid == 0) {
    float s = 0.0f;
#pragma unroll
    for (int i = 0; i < WPB; ++i) s += shW[i];
    ws[WS_PART + blockIdx.x] = s;
  }
}

// ---------------------------------------------------------------------------
// Final: fixed-order tree reduction of the 256 partials (deterministic),
// result = -(log(sum) - C0)/N.
// ---------------------------------------------------------------------------
__global__ __launch_bounds__(256) void gmm_final(const float* __restrict__ ws,
                                                 float* __restrict__ out) {
  __shared__ float sh[256];
  const int tid = threadIdx.x;
  sh[tid] = ws[WS_PART + tid];
  __syncthreads();
  for (int s = 128; s > 0; s >>= 1) {
    if (tid < s) sh[tid] += sh[tid + s];
    __syncthreads();
  }
  if (tid == 0) {
    float C0 = ws[WS_C0];
    out[0] = -(logf(sh[0]) - C0) / (float)NPTS;
  }
}

extern "C" void kernel_launch(void* const* d_in, const int* in_sizes, int n_in,
                              void* d_out, int out_size, void* d_ws, size_t ws_size,
                              hipStream_t stream) {
  const float* X       = (const float*)d_in[0];
  const float* means   = (const float*)d_in[1];
  const float* chols   = (const float*)d_in[2];
  const float* weights = (const float*)d_in[3];
  // d_in[4] = it (unused by the math)
  float* ws  = (float*)d_ws;
  float* out = (float*)d_out;
  (void)in_sizes; (void)n_in; (void)out_size; (void)ws_size;

  gmm_setup<<<1, 32, 0, stream>>>(means, chols, weights, ws);
  gmm_main<<<NBLK, 256, 0, stream>>>(X, ws);
  gmm_final<<<1, 256, 0, stream>>>(ws, out);
}